// DSS_69853348102186
// MI455X (gfx1250) — compile-verified
//
#include <hip/hip_runtime.h>

#define B_ 2
#define L_ 8192
#define D_ 1024
#define N_ 512
#define LC 512              // chunk length
#define NCH (L_ / LC)       // 16 chunks
#define EPSF 1e-5f

typedef float v2f __attribute__((ext_vector_type(2)));
typedef float v8f __attribute__((ext_vector_type(8)));

#define WMMA_F32(A, Bv, Cv) \
  __builtin_amdgcn_wmma_f32_16x16x4_f32(false, (A), false, (Bv), (short)0, (Cv), false, false)

// ---------------- LayerNorm: u = (x-mean)*rsqrt(var+eps)*gamma + beta ------
__global__ void ln_kernel(const float* __restrict__ x,
                          const float* __restrict__ gamma,
                          const float* __restrict__ beta,
                          float* __restrict__ u) {
  const int row = blockIdx.x;            // b*L + l
  const long base = (long)row * D_;
  __shared__ float s0[256];
  __shared__ float s1[256];
  float v[4];
  float sum = 0.f, sq = 0.f;
#pragma unroll
  for (int k = 0; k < 4; ++k) {
    v[k] = x[base + threadIdx.x + k * 256];
    sum += v[k];
    sq  += v[k] * v[k];
  }
  s0[threadIdx.x] = sum;
  s1[threadIdx.x] = sq;
  __syncthreads();
  for (int off = 128; off > 0; off >>= 1) {
    if ((int)threadIdx.x < off) {
      s0[threadIdx.x] += s0[threadIdx.x + off];
      s1[threadIdx.x] += s1[threadIdx.x + off];
    }
    __syncthreads();
  }
  const float mean = s0[0] * (1.0f / D_);
  const float var  = s1[0] * (1.0f / D_) - mean * mean;
  const float inv  = rsqrtf(var + EPSF);
#pragma unroll
  for (int k = 0; k < 4; ++k) {
    const int d = threadIdx.x + k * 256;
    u[base + d] = (v[k] - mean) * inv * gamma[d] + beta[d];
  }
}

// ------- S[l, 0:N]=Re exp(Lam*l), S[l, N:2N]=Im exp(Lam*l), l < LC ---------
__global__ void sgen_kernel(const float* __restrict__ lr,
                            const float* __restrict__ li,
                            float* __restrict__ S) {
  const int idx = blockIdx.x * 256 + threadIdx.x;
  if (idx >= LC * N_) return;
  const int l = idx >> 9;            // / N_
  const int n = idx & (N_ - 1);
  const float a = -expf(lr[n]);
  const float b =  expf(li[n]);
  const float t = (float)l;
  const float e = expf(a * t);
  float sn, cs;
  sincosf(b * t, &sn, &cs);
  S[(long)l * (2 * N_) + n]      = e * cs;
  S[(long)l * (2 * N_) + N_ + n] = e * sn;
}

// ------ C' = (Cr+iCi)(e^Lam-1)/Lam; Cf=[Re C'; -Im C'] (2N,D); Cp (N,D) ----
__global__ void cgen_kernel(const float* __restrict__ lr,
                            const float* __restrict__ li,
                            const float* __restrict__ Cr,
                            const float* __restrict__ Ci,
                            float* __restrict__ Cf,
                            float* __restrict__ Cpr,
                            float* __restrict__ Cpi) {
  const int idx = blockIdx.x * 256 + threadIdx.x;
  if (idx >= N_ * D_) return;
  const int n = idx >> 10;           // / D_
  const int d = idx & (D_ - 1);
  const float a = -expf(lr[n]);
  const float b =  expf(li[n]);
  const float ea = expf(a);
  float sn, cs;
  sincosf(b, &sn, &cs);
  const float er = ea * cs - 1.0f;
  const float ei = ea * sn;
  const float den = 1.0f / (a * a + b * b);
  const float wr = (er * a + ei * b) * den;
  const float wi = (ei * a - er * b) * den;
  const float cr = Cr[(long)d * N_ + n];
  const float ci = Ci[(long)d * N_ + n];
  const float cpr = cr * wr - ci * wi;
  const float cpi = cr * wi + ci * wr;
  Cf[(long)n * D_ + d]        =  cpr;
  Cf[(long)(N_ + n) * D_ + d] = -cpi;
  Cpr[(long)n * D_ + d] = cpr;
  Cpi[(long)n * D_ + d] = cpi;
}

// ---- P[j,:]=[Re e^{Lam(j+1)} | Im], Q[n,t]=e^{Lam(LC-1-t)}, E=e^{Lam*LC} --
__global__ void pqe_kernel(const float* __restrict__ lr,
                           const float* __restrict__ li,
                           float* __restrict__ Pf,   // [LC][2N]
                           float* __restrict__ Qr,   // [N][LC]
                           float* __restrict__ Qi,   // [N][LC]
                           float* __restrict__ Ev) { // er[0:N], ei[N:2N]
  const int idx = blockIdx.x * 256 + threadIdx.x;
  if (idx >= N_ * LC) return;
  const int n = idx >> 9;            // / LC
  const int t = idx & (LC - 1);
  const float a = -expf(lr[n]);
  const float b =  expf(li[n]);
  {
    const float ja = (float)(t + 1);
    const float e = expf(a * ja);
    float sn, cs;
    sincosf(b * ja, &sn, &cs);
    Pf[(long)t * (2 * N_) + n]      = e * cs;
    Pf[(long)t * (2 * N_) + N_ + n] = e * sn;
  }
  {
    const float qa = (float)(LC - 1 - t);
    const float e = expf(a * qa);
    float sn, cs;
    sincosf(b * qa, &sn, &cs);
    Qr[(long)n * LC + t] = e * cs;
    Qi[(long)n * LC + t] = e * sn;
  }
  if (t == 0) {
    const float ca = (float)LC;
    const float e = expf(a * ca);
    float sn, cs;
    sincosf(b * ca, &sn, &cs);
    Ev[n]      = e * cs;
    Ev[N_ + n] = e * sn;
  }
}

__global__ void hzero_kernel(float* __restrict__ hr, float* __restrict__ hi) {
  const long idx = (long)blockIdx.x * 256 + threadIdx.x;
  if (idx >= (long)B_ * N_ * D_) return;
  hr[idx] = 0.f;
  hi[idx] = 0.f;
}

// ---------------- K[0:LC] = S @ Cf  via f32 WMMA 16x16x4 -------------------
__global__ void kgemm_kernel(const float* __restrict__ S,
                             const float* __restrict__ Cf,
                             float* __restrict__ K) {
  const int lane = threadIdx.x & 31;
  const int wave = threadIdx.x >> 5;
  const int m  = lane & 15;
  const int hb = lane >> 4;
  const int kb = hb * 2;
  const int l0 = blockIdx.y * 128 + wave * 16;
  const int d0 = blockIdx.x * 64;

  v8f a0 = {}, a1 = {}, a2 = {}, a3 = {};
  const float* Sr = S  + (long)(l0 + m) * (2 * N_) + kb;
  const float* Bb = Cf + (long)kb * D_ + d0 + m;
  for (int kk = 0; kk < 2 * N_; kk += 4) {
    v2f a; a.x = Sr[kk]; a.y = Sr[kk + 1];
    const float* bp = Bb + (long)kk * D_;
    v2f b0, b1, b2, b3;
    b0.x = bp[0];  b0.y = bp[D_];
    b1.x = bp[16]; b1.y = bp[D_ + 16];
    b2.x = bp[32]; b2.y = bp[D_ + 32];
    b3.x = bp[48]; b3.y = bp[D_ + 48];
    a0 = WMMA_F32(a, b0, a0);
    a1 = WMMA_F32(a, b1, a1);
    a2 = WMMA_F32(a, b2, a2);
    a3 = WMMA_F32(a, b3, a3);
  }
#pragma unroll
  for (int r = 0; r < 8; ++r) {
    const long o = (long)(l0 + r + 8 * hb) * D_ + d0 + m;
    K[o] = a0[r]; K[o + 16] = a1[r]; K[o + 32] = a2[r]; K[o + 48] = a3[r];
  }
}

// ------------- intra-chunk causal conv + residual (t < LC only) ------------
__global__ void conv_kernel(const float* __restrict__ u,
                            const float* __restrict__ K,
                            const float* __restrict__ pD,
                            float* __restrict__ y) {
  __shared__ float Kt[64 * 64];
  __shared__ float Ut[128 * 64];
  const int dt = threadIdx.x & 63;
  const int lt = threadIdx.x >> 6;
  const int d0 = blockIdx.x * 64;
  const int l0 = blockIdx.y * 64;
  const int b  = blockIdx.z;
  const int d  = d0 + dt;
  const long bb = (long)b * L_ * D_;
  const int cl = l0 & ~(LC - 1);          // chunk start

  float acc[16];
#pragma unroll
  for (int j = 0; j < 16; ++j) acc[j] = 0.f;

  const int ntb = ((l0 - cl) >> 6) + 1;   // <= LC/64 = 8
  for (int tb = 0; tb < ntb; ++tb) {
    const int t0 = tb * 64;
#pragma unroll
    for (int rr = 0; rr < 16; ++rr) {
      const int r = lt * 16 + rr;
      Kt[r * 64 + dt] = K[(long)(t0 + r) * D_ + d];
    }
    const int gb = l0 - t0 - 63;
#pragma unroll
    for (int rr = 0; rr < 32; ++rr) {
      const int r = lt * 32 + rr;
      const int g = gb + r;
      Ut[r * 64 + dt] = (g >= cl && g < L_) ? u[bb + (long)g * D_ + d] : 0.f;
    }
    __syncthreads();
#pragma unroll 4
    for (int tp = 0; tp < 64; ++tp) {
      const float kv = Kt[tp * 64 + dt];
#pragma unroll
      for (int j = 0; j < 16; ++j) {
        const int li = j * 4 + lt;
        acc[j] = fmaf(kv, Ut[(li + 63 - tp) * 64 + dt], acc[j]);
      }
    }
    __syncthreads();
  }

  const float pd = pD[d];
#pragma unroll
  for (int j = 0; j < 16; ++j) {
    const long o = bb + (long)(l0 + j * 4 + lt) * D_ + d;
    y[o] = acc[j] + u[o] * pd;
  }
}

// --------- fused per-chunk phase: yacc (role 0) + zupd (role 1) ------------
// role 0: y[chunk c] += Pf @ Wcur[b]            (skipped for c==0)
// role 1: h <- E*h + Q @ u[chunk c];  Wnext = C' o h_new   (skipped c==NCH-1)
__global__ void phase_kernel(const float* __restrict__ u,
                             const float* __restrict__ Pf,
                             const float* __restrict__ Qr,
                             const float* __restrict__ Qi,
                             const float* __restrict__ Ev,
                             const float* __restrict__ Cpr,
                             const float* __restrict__ Cpi,
                             const float* __restrict__ Wcur,
                             float* __restrict__ Wnext,
                             float* __restrict__ hr,
                             float* __restrict__ hi,
                             float* __restrict__ y,
                             int c) {
  const int lane = threadIdx.x & 31;
  const int wave = threadIdx.x >> 5;
  const int m    = lane & 15;
  const int hb   = lane >> 4;
  const int kb   = hb * 2;
  const int d0   = blockIdx.x * 32;                 // 32-wide d tile
  const int row0 = blockIdx.y * 128 + wave * 16;    // j0 (yacc) or n0 (zupd)
  const int role = blockIdx.z >> 1;
  const int b    = blockIdx.z & 1;

  if (role == 0) {
    if (c == 0) return;
    v8f accA = {}, accB = {};
    const float* Ap = Pf + (long)(row0 + m) * (2 * N_) + kb;
    const float* Bb = Wcur + (long)b * 2 * N_ * D_ + (long)kb * D_ + d0 + m;
    for (int kk = 0; kk < 2 * N_; kk += 4) {
      v2f a; a.x = Ap[kk]; a.y = Ap[kk + 1];
      const float* bp = Bb + (long)kk * D_;
      v2f b0, b1;
      b0.x = bp[0];  b0.y = bp[D_];
      b1.x = bp[16]; b1.y = bp[D_ + 16];
      accA = WMMA_F32(a, b0, accA);
      accB = WMMA_F32(a, b1, accB);
    }
#pragma unroll
    for (int r = 0; r < 8; ++r) {
      const int l = c * LC + row0 + r + 8 * hb;
      const long o = ((long)b * L_ + l) * D_ + d0 + m;
      y[o]      += accA[r];
      y[o + 16] += accB[r];
    }
  } else {
    if (c == NCH - 1) return;
    v8f zr0 = {}, zr1 = {}, zi0 = {}, zi1 = {};
    const float* Ar = Qr + (long)(row0 + m) * LC + kb;
    const float* Ai = Qi + (long)(row0 + m) * LC + kb;
    const float* Bb = u + ((long)b * L_ + (long)c * LC) * D_ + (long)kb * D_ + d0 + m;
    for (int kk = 0; kk < LC; kk += 4) {
      v2f ar; ar.x = Ar[kk]; ar.y = Ar[kk + 1];
      v2f ai; ai.x = Ai[kk]; ai.y = Ai[kk + 1];
      const float* bp = Bb + (long)kk * D_;
      v2f b0, b1;
      b0.x = bp[0];  b0.y = bp[D_];
      b1.x = bp[16]; b1.y = bp[D_ + 16];
      zr0 = WMMA_F32(ar, b0, zr0);  zi0 = WMMA_F32(ai, b0, zi0);
      zr1 = WMMA_F32(ar, b1, zr1);  zi1 = WMMA_F32(ai, b1, zi1);
    }
    const long wb = (long)b * 2 * N_ * D_;
#pragma unroll
    for (int r = 0; r < 8; ++r) {
      const int n = row0 + r + 8 * hb;
      const float er = Ev[n], ei = Ev[N_ + n];
      const long cbase = (long)n * D_ + d0 + m;
      const long hbase = (long)b * N_ * D_ + cbase;
      {
        const float R = hr[hbase], I = hi[hbase];
        const float nR = er * R - ei * I + zr0[r];
        const float nI = er * I + ei * R + zi0[r];
        hr[hbase] = nR;
        hi[hbase] = nI;
        const float cr = Cpr[cbase], ci = Cpi[cbase];
        Wnext[wb + cbase]                 =  cr * nR - ci * nI;
        Wnext[wb + (long)N_ * D_ + cbase] = -(cr * nI + ci * nR);
      }
      {
        const float R = hr[hbase + 16], I = hi[hbase + 16];
        const float nR = er * R - ei * I + zr1[r];
        const float nI = er * I + ei * R + zi1[r];
        hr[hbase + 16] = nR;
        hi[hbase + 16] = nI;
        const float cr = Cpr[cbase + 16], ci = Cpi[cbase + 16];
        Wnext[wb + cbase + 16]                 =  cr * nR - ci * nI;
        Wnext[wb + (long)N_ * D_ + cbase + 16] = -(cr * nI + ci * nR);
      }
    }
  }
}

// ---------------------------------------------------------------------------
extern "C" void kernel_launch(void* const* d_in, const int* in_sizes, int n_in,
                              void* d_out, int out_size, void* d_ws, size_t ws_size,
                              hipStream_t stream) {
  (void)in_sizes; (void)n_in; (void)out_size; (void)ws_size;
  const float* x  = (const float*)d_in[0];
  const float* lr = (const float*)d_in[1];
  const float* li = (const float*)d_in[2];
  const float* Cr = (const float*)d_in[3];
  const float* Ci = (const float*)d_in[4];
  const float* pD = (const float*)d_in[5];
  const float* g  = (const float*)d_in[6];
  const float* be = (const float*)d_in[7];
  float* out = (float*)d_out;

  float* ws = (float*)d_ws;
  size_t o = 0;
  float* u   = ws + o; o += (size_t)B_ * L_ * D_;      // 64 MB
  float* S   = ws + o; o += (size_t)LC * 2 * N_;       //  2 MB
  float* Cf  = ws + o; o += (size_t)2 * N_ * D_;       //  4 MB
  float* Cpr = ws + o; o += (size_t)N_ * D_;           //  2 MB
  float* Cpi = ws + o; o += (size_t)N_ * D_;           //  2 MB
  float* K   = ws + o; o += (size_t)LC * D_;           //  2 MB
  float* Pf  = ws + o; o += (size_t)LC * 2 * N_;       //  2 MB
  float* Qr  = ws + o; o += (size_t)N_ * LC;           //  1 MB
  float* Qi  = ws + o; o += (size_t)N_ * LC;           //  1 MB
  float* Ev  = ws + o; o += (size_t)2 * N_;            //  4 KB
  float* hr  = ws + o; o += (size_t)B_ * N_ * D_;      //  4 MB
  float* hi  = ws + o; o += (size_t)B_ * N_ * D_;      //  4 MB
  float* W0  = ws + o; o += (size_t)B_ * 2 * N_ * D_;  //  8 MB
  float* W1  = ws + o; o += (size_t)B_ * 2 * N_ * D_;  //  8 MB

  // setup (independent)
  ln_kernel   <<<B_ * L_, 256, 0, stream>>>(x, g, be, u);
  sgen_kernel <<<(LC * N_) / 256, 256, 0, stream>>>(lr, li, S);
  cgen_kernel <<<(N_ * D_) / 256, 256, 0, stream>>>(lr, li, Cr, Ci, Cf, Cpr, Cpi);
  pqe_kernel  <<<(N_ * LC) / 256, 256, 0, stream>>>(lr, li, Pf, Qr, Qi, Ev);
  hzero_kernel<<<(B_ * N_ * D_) / 256, 256, 0, stream>>>(hr, hi);
  kgemm_kernel<<<dim3(D_ / 64, LC / 128), 256, 0, stream>>>(S, Cf, K);
  // intra-chunk conv + residual writes y
  conv_kernel<<<dim3(D_ / 64, L_ / 64, B_), 256, 0, stream>>>(u, K, pD, out);

  // sequential chunk scan: one fused kernel per chunk
  // zupd(c) writes W[(c+1)&1]; yacc(c) reads W[c&1]
  float* Wb[2] = {W0, W1};
  for (int c = 0; c < NCH; ++c) {
    phase_kernel<<<dim3(D_ / 32, LC / 128, 4), 256, 0, stream>>>(
        u, Pf, Qr, Qi, Ev, Cpr, Cpi,
        Wb[c & 1], Wb[(c + 1) & 1], hr, hi, out, c);
  }
}